// GCN_4973572128804
// MI455X (gfx1250) — compile-verified
//
#include <hip/hip_runtime.h>

// ---------------------------------------------------------------------------
// GCN forward on MI455X (gfx1250): 3x [ WMMA-f32 GEMM -> bias-init ->
// edge scatter-add (L2-resident atomics) -> ReLU ].
// fp32 matrix path: V_WMMA_F32_16X16X4_F32 (full reference precision).
// GEMM uses 2x2 register tiling per wave: 4 WMMAs per K-step sharing
// A/B operand loads (2 loads per WMMA instead of 4).
// ---------------------------------------------------------------------------

typedef float v2f __attribute__((ext_vector_type(2)));
typedef float v8f __attribute__((ext_vector_type(8)));

#define GCN_IN_C  256
#define GCN_H1    128
#define GCN_H2    64
#define GCN_OUT   64

// ----------------------------------------------------------------------
// Dense GEMM  C[M,N] = A[M,K] @ W[K,N] using fp32 WMMA 16x16x4.
// One wave computes a 32x32 patch of C (2x2 tiles of 16x16).
// blockDim.x = (N/32)*32, grid.x = M/32 (M=100000 -> 3125 blocks).
//
// VGPR layouts (CDNA5 ISA 7.12.2):
//  A 16x4 f32 : lanes 0-15 -> K={0,1} in v0/v1 ; lanes 16-31 -> K={2,3}
//  B 4x16 f32 : row striped across lanes; half-wave split mirrors A
//  C/D 16x16  : VGPR v -> M = v + 8*(lane/16), N = lane%16
// ----------------------------------------------------------------------
template <int K, int N>
__global__ __launch_bounds__(128)
void gcn_gemm_wmma_f32(const float* __restrict__ A,
                       const float* __restrict__ W,
                       float* __restrict__ C,
                       int M)
{
    const int lane = threadIdx.x & 31;
    const int wave = threadIdx.x >> 5;      // 32-col slab index within block
    const int m0   = blockIdx.x << 5;       // 32 rows per block
    const int n0   = wave << 5;             // 32 cols per wave
    const int half = lane >> 4;             // 0: K pair {0,1}, 1: K pair {2,3}
    const int l    = lane & 15;

    // A rows for the two M tiles, pre-offset by the half-wave K pair.
    const float* __restrict__ arow0 = A + (size_t)(m0 + l) * K + 2 * half;
    const float* __restrict__ arow1 = arow0 + (size_t)16 * K;
    // B columns for the two N tiles, pre-offset by the half-wave K pair.
    const float* __restrict__ wcol0 = W + (size_t)(2 * half) * N + (n0 + l);
    const float* __restrict__ wcol1 = wcol0 + 16;

    v8f acc00 = {}, acc01 = {}, acc10 = {}, acc11 = {};
#pragma unroll 2
    for (int k = 0; k < K; k += 4) {
        v2f a0, a1, b0, b1;
        a0.x = arow0[k + 0];                 // K = k + 2*half     (b64 pair)
        a0.y = arow0[k + 1];
        a1.x = arow1[k + 0];
        a1.y = arow1[k + 1];
        b0.x = wcol0[(size_t)k * N];         // row (k + 2*half)
        b0.y = wcol0[(size_t)k * N + N];     // row (k + 2*half + 1)
        b1.x = wcol1[(size_t)k * N];
        b1.y = wcol1[(size_t)k * N + N];
        // (neg_a, A, neg_b, B, c_mod, C, reuse_a, reuse_b)
        acc00 = __builtin_amdgcn_wmma_f32_16x16x4_f32(
            false, a0, false, b0, (short)0, acc00, false, false);
        acc01 = __builtin_amdgcn_wmma_f32_16x16x4_f32(
            false, a0, false, b1, (short)0, acc01, false, false);
        acc10 = __builtin_amdgcn_wmma_f32_16x16x4_f32(
            false, a1, false, b0, (short)0, acc10, false, false);
        acc11 = __builtin_amdgcn_wmma_f32_16x16x4_f32(
            false, a1, false, b1, (short)0, acc11, false, false);
    }

    // Store: within a tile, VGPR v holds M = mtile + v + 8*half, N = ntile + l.
    float* __restrict__ crow0 = C + (size_t)(m0 + 8 * half) * N + (n0 + l);
    float* __restrict__ crow1 = crow0 + (size_t)16 * N;
#pragma unroll
    for (int v = 0; v < 8; ++v) {
        crow0[(size_t)v * N]      = acc00[v];
        crow0[(size_t)v * N + 16] = acc01[v];
        crow1[(size_t)v * N]      = acc10[v];
        crow1[(size_t)v * N + 16] = acc11[v];
    }
}

// ----------------------------------------------------------------------
// agg[n, :] starts as bias; edges scatter-add sup[src]*w into agg[dst].
// Consecutive threads cover consecutive 4-channel chunks of one edge:
// coalesced float4 gather from the (L2-resident) support row, then four
// global_atomic_add_f32 into the destination row.
// ----------------------------------------------------------------------
template <int C>
__global__ __launch_bounds__(256)
void gcn_edge_scatter(const float* __restrict__ sup,
                      const int* __restrict__ src,
                      const int* __restrict__ dst,
                      const float* __restrict__ ew,
                      float* __restrict__ agg,
                      int E)
{
    constexpr int CHUNKS = C / 4;
    long long tid = (long long)blockIdx.x * blockDim.x + threadIdx.x;
    if (tid >= (long long)E * CHUNKS) return;
    const int e = (int)(tid / CHUNKS);
    const int c = (int)(tid % CHUNKS) * 4;

    const int   s = src[e];
    const int   d = dst[e];
    const float w = ew[e];

    const float4 v = *reinterpret_cast<const float4*>(sup + (size_t)s * C + c);
    float* o = agg + (size_t)d * C + c;
    atomicAdd(o + 0, v.x * w);
    atomicAdd(o + 1, v.y * w);
    atomicAdd(o + 2, v.z * w);
    atomicAdd(o + 3, v.w * w);
}

// agg[n*C + c] = b[c], float4-vectorized (C is a power of two, >= 64).
__global__ __launch_bounds__(256)
void gcn_fill_bias(float* __restrict__ out, const float* __restrict__ b,
                   int C, long long total4)
{
    long long i = (long long)blockIdx.x * blockDim.x + threadIdx.x;
    if (i >= total4) return;
    const int c = (int)((i * 4) & (long long)(C - 1));
    const float4 v = *reinterpret_cast<const float4*>(b + c);
    *reinterpret_cast<float4*>(out + i * 4) = v;
}

__global__ __launch_bounds__(256)
void gcn_relu(float* __restrict__ x, long long total4)
{
    long long i = (long long)blockIdx.x * blockDim.x + threadIdx.x;
    if (i >= total4) return;
    float4 v = *reinterpret_cast<float4*>(x + i * 4);
    v.x = fmaxf(v.x, 0.0f);
    v.y = fmaxf(v.y, 0.0f);
    v.z = fmaxf(v.z, 0.0f);
    v.w = fmaxf(v.w, 0.0f);
    *reinterpret_cast<float4*>(x + i * 4) = v;
}

// ---------------------------------------------------------------------------
extern "C" void kernel_launch(void* const* d_in, const int* in_sizes, int n_in,
                              void* d_out, int out_size, void* d_ws, size_t ws_size,
                              hipStream_t stream)
{
    const float* x    = (const float*)d_in[0];
    const int*   esrc = (const int*)  d_in[1];
    const int*   edst = (const int*)  d_in[2];
    const float* ew   = (const float*)d_in[3];
    const float* W1   = (const float*)d_in[4];
    const float* b1   = (const float*)d_in[5];
    const float* W2   = (const float*)d_in[6];
    const float* b2   = (const float*)d_in[7];
    const float* W3   = (const float*)d_in[8];
    const float* b3   = (const float*)d_in[9];

    const int M = in_sizes[0] / GCN_IN_C;   // 100000 (multiple of 32)
    const int E = in_sizes[1];              // 1600000

    // Workspace carve-out (reuses sup128 for layer-3 support).
    float* ws     = (float*)d_ws;
    float* sup128 = ws;                                   // M*128: sup1, sup3
    float* h128   = sup128 + (size_t)M * GCN_H1;          // M*128: h1
    float* sup64  = h128   + (size_t)M * GCN_H1;          // M*64 : sup2
    float* h64    = sup64  + (size_t)M * GCN_H2;          // M*64 : h2
    float* out    = (float*)d_out;                        // M*64

    const int mt = M / 32;                  // 32-row patches per GEMM block
    const long long q128 = (long long)M * GCN_H1 / 4;   // float4 counts
    const long long q64  = (long long)M * GCN_H2 / 4;
    const long long e128 = (long long)E * (GCN_H1 / 4);
    const long long e64  = (long long)E * (GCN_H2 / 4);
    const int B = 256;

    // ---- Layer 1: h1 = relu(A @ (x @ W1) + b1) ----
    gcn_gemm_wmma_f32<GCN_IN_C, GCN_H1>
        <<<mt, (GCN_H1 / 32) * 32, 0, stream>>>(x, W1, sup128, M);
    gcn_fill_bias<<<(int)((q128 + B - 1) / B), B, 0, stream>>>(h128, b1, GCN_H1, q128);
    gcn_edge_scatter<GCN_H1>
        <<<(int)((e128 + B - 1) / B), B, 0, stream>>>(sup128, esrc, edst, ew, h128, E);
    gcn_relu<<<(int)((q128 + B - 1) / B), B, 0, stream>>>(h128, q128);

    // ---- Layer 2: h2 = relu(A @ (h1 @ W2) + b2) ----
    gcn_gemm_wmma_f32<GCN_H1, GCN_H2>
        <<<mt, (GCN_H2 / 32) * 32, 0, stream>>>(h128, W2, sup64, M);
    gcn_fill_bias<<<(int)((q64 + B - 1) / B), B, 0, stream>>>(h64, b2, GCN_H2, q64);
    gcn_edge_scatter<GCN_H2>
        <<<(int)((e64 + B - 1) / B), B, 0, stream>>>(sup64, esrc, edst, ew, h64, E);
    gcn_relu<<<(int)((q64 + B - 1) / B), B, 0, stream>>>(h64, q64);

    // ---- Layer 3: out = A @ (h2 @ W3) + b3 ----
    gcn_gemm_wmma_f32<GCN_H2, GCN_OUT>
        <<<mt, (GCN_OUT / 32) * 32, 0, stream>>>(h64, W3, sup128, M);
    gcn_fill_bias<<<(int)((q64 + B - 1) / B), B, 0, stream>>>(out, b3, GCN_OUT, q64);
    gcn_edge_scatter<GCN_OUT>
        <<<(int)((e64 + B - 1) / B), B, 0, stream>>>(sup128, esrc, edst, ew, out, E);
}